// RoiProposal_5394478923802
// MI455X (gfx1250) — compile-verified
//
#include <hip/hip_runtime.h>
#include <hip/hip_bf16.h>
#include <cstdint>

// ---------------- problem constants ----------------
#define RPN_H      64
#define RPN_W      64
#define RPN_A      9
#define NANCH      (RPN_H * RPN_W * RPN_A)   // 36864
#define SORT_N     65536                     // next pow2 >= NANCH
#define KNMS       6000                      // PRE_NMS_TOPN (< NANCH)
#define WORDS      188                       // ceil(KNMS/32)
#define KPAD       (WORDS * 32)              // 6016
#define ROWT       (KNMS / 16)               // 375 row tiles
#define POST       300
#define NMS_T      0.7f
#define IMW1       1023.0f
#define IMH1       1023.0f
#define MINSZ      16.0f

// workspace layout (bytes, all 256-aligned; total ~5.75 MB)
#define OFF_BOXES  0u                        // 36864 * 4 f32
#define OFF_KEY    589824u                   // 65536 f32
#define OFF_VAL    851968u                   // 65536 i32
#define OFF_TOPB   1114112u                  // 6016 * 4 f32
#define OFF_TOPS   1210368u                  // 6000 f32
#define OFF_MASK   1234432u                  // 6000 * 188 u32
#define OFF_KEEPW  5746432u                  // 188 u32

typedef __attribute__((ext_vector_type(2))) float v2f;
typedef __attribute__((ext_vector_type(8))) float v8f;

__device__ __forceinline__ uint32_t ballot32(bool p) {
#if defined(__has_builtin)
#if __has_builtin(__builtin_amdgcn_ballot_w32)
  return __builtin_amdgcn_ballot_w32(p);
#else
  return (uint32_t)__ballot(p);
#endif
#else
  return (uint32_t)__ballot(p);
#endif
}

// ---------------- 1) anchor decode + score ----------------
__global__ void rpn_decode_kernel(const float* __restrict__ cls,
                                  const float* __restrict__ bb,
                                  float* __restrict__ boxes,
                                  float* __restrict__ key,
                                  int* __restrict__ val) {
  int n = blockIdx.x * blockDim.x + threadIdx.x;
  if (n >= SORT_N) return;
  val[n] = n;
  if (n >= NANCH) { key[n] = -INFINITY; return; }

  int a = n % RPN_A;
  int cell = n / RPN_A;
  int x = cell % RPN_W;
  int y = cell / RPN_W;
  int ridx = a / 3, sidx = a % 3;
  // py-faster-rcnn base anchors (base 16, ratios .5/1/2 with banker's rounding)
  const float W0[3] = {23.f, 16.f, 11.f};
  const float H0[3] = {12.f, 16.f, 22.f};
  float scale = (float)(8 << sidx);
  float aw = W0[ridx] * scale;
  float ah = H0[ridx] * scale;
  float ax1 = x * 16.f + 7.5f - 0.5f * (aw - 1.f);
  float ay1 = y * 16.f + 7.5f - 0.5f * (ah - 1.f);
  float axc = ax1 + 0.5f * aw;
  float ayc = ay1 + 0.5f * ah;

  int cbase = (y * RPN_W + x) * (2 * RPN_A) + 2 * a;
  float l0 = cls[cbase], l1 = cls[cbase + 1];
  float score = 1.f / (1.f + __expf(l0 - l1) * 0.f + expf(l0 - l1) * 0.f + expf(l0 - l1));
  // (kept as plain expf; the redundant zero-terms fold away)
  score = 1.f / (1.f + expf(l0 - l1));

  int dbase = (y * RPN_W + x) * (4 * RPN_A) + 4 * a;
  float dx = bb[dbase + 0], dy = bb[dbase + 1];
  float dw = bb[dbase + 2], dh = bb[dbase + 3];
  float pxc = dx * aw + axc;
  float pyc = dy * ah + ayc;
  float pw = expf(dw) * aw;
  float ph = expf(dh) * ah;
  float x1 = pxc - 0.5f * pw, y1 = pyc - 0.5f * ph;
  float x2 = pxc + 0.5f * pw, y2 = pyc + 0.5f * ph;
  x1 = fminf(fmaxf(x1, 0.f), IMW1);
  x2 = fminf(fmaxf(x2, 0.f), IMW1);
  y1 = fminf(fmaxf(y1, 0.f), IMH1);
  y2 = fminf(fmaxf(y2, 0.f), IMH1);
  boxes[n * 4 + 0] = x1; boxes[n * 4 + 1] = y1;
  boxes[n * 4 + 2] = x2; boxes[n * 4 + 3] = y2;
  bool valid = (x2 - x1 + 1.f >= MINSZ) && (y2 - y1 + 1.f >= MINSZ);
  key[n] = valid ? score : -INFINITY;
}

// ---------------- 2) bitonic sort (desc score, asc index tiebreak) ----------------
__global__ void bitonic_kernel(float* __restrict__ key, int* __restrict__ val,
                               int j, int k) {
  int i = blockIdx.x * blockDim.x + threadIdx.x;
  int ixj = i ^ j;
  if (ixj <= i) return;
  float ki = key[i], kj = key[ixj];
  int vi = val[i], vj = val[ixj];
  bool desc = ((i & k) == 0);
  bool swap = desc ? ((ki < kj) || (ki == kj && vi > vj))
                   : ((ki > kj) || (ki == kj && vi < vj));
  if (swap) { key[i] = kj; key[ixj] = ki; val[i] = vj; val[ixj] = vi; }
}

// ---------------- 3) gather top-K boxes ----------------
__global__ void gather_kernel(const float* __restrict__ key, const int* __restrict__ val,
                              const float* __restrict__ boxes,
                              float* __restrict__ topb, float* __restrict__ tops) {
  int t = blockIdx.x * blockDim.x + threadIdx.x;
  if (t >= KPAD) return;
  if (t < KNMS) {
    int n = val[t];
    topb[t * 4 + 0] = boxes[n * 4 + 0];
    topb[t * 4 + 1] = boxes[n * 4 + 1];
    topb[t * 4 + 2] = boxes[n * 4 + 2];
    topb[t * 4 + 3] = boxes[n * 4 + 3];
    tops[t] = key[t];
  } else {
    topb[t * 4 + 0] = 0.f; topb[t * 4 + 1] = 0.f;
    topb[t * 4 + 2] = 0.f; topb[t * 4 + 3] = 0.f;
  }
}

// ---------------- 4) NMS suppression bitmask ----------------
// One wave32 per 16(row)x32(col) tile. area_i + area_j computed on the matrix
// pipe via V_WMMA_F32_16X16X4_F32 rank-2 outer product: A = [area_i, 1],
// B = [1 ; area_j]. C layout (lane%16 = N, vgpr+8*(lane/16) = M) matches
// ballot bit order, so each 16-bit half-ballot is one row's suppression bits.
__global__ void __launch_bounds__(256) nms_mask_kernel(const float* __restrict__ tb,
                                                       uint32_t* __restrict__ mask) {
  const int lane = threadIdx.x & 31;
  const int wv = threadIdx.x >> 5;
  const int gw = blockIdx.x * (blockDim.x >> 5) + wv;  // wave-uniform
  const int total = ROWT * WORDS;
  if (gw < total) {
    const int rt = gw / WORDS;
    const int wc = gw % WORDS;
    const int i0 = rt * 16;
    const int j0 = wc * 32;

    const float4 rb = reinterpret_cast<const float4*>(tb)[i0 + (lane & 15)];
    const float4 cb = reinterpret_cast<const float4*>(tb)[j0 + lane];
    const float ar = (rb.z - rb.x + 1.f) * (rb.w - rb.y + 1.f);
    const float ac = (cb.z - cb.x + 1.f) * (cb.w - cb.y + 1.f);
    const float ac_hi = __shfl(ac, lane + 16, 32);
    const bool lo = lane < 16;

    v2f A;  A.x  = lo ? ar : 0.f;    A.y  = lo ? 1.f : 0.f;
    v2f B0; B0.x = lo ? 1.f : 0.f;   B0.y = lo ? ac : 0.f;
    v2f B1; B1.x = lo ? 1.f : 0.f;   B1.y = lo ? ac_hi : 0.f;
    v8f c0 = {}; v8f c1 = {};
    c0 = __builtin_amdgcn_wmma_f32_16x16x4_f32(false, A, false, B0, (short)0, c0, false, false);
    c1 = __builtin_amdgcn_wmma_f32_16x16x4_f32(false, A, false, B1, (short)0, c1, false, false);

    const int nsrc = lane & 15;
    const float c0x1 = __shfl(cb.x, nsrc, 32),      c0y1 = __shfl(cb.y, nsrc, 32);
    const float c0x2 = __shfl(cb.z, nsrc, 32),      c0y2 = __shfl(cb.w, nsrc, 32);
    const float c1x1 = __shfl(cb.x, nsrc + 16, 32), c1y1 = __shfl(cb.y, nsrc + 16, 32);
    const float c1x2 = __shfl(cb.z, nsrc + 16, 32), c1y2 = __shfl(cb.w, nsrc + 16, 32);

    const int mhalf = (lane >> 4) << 3;  // this lane covers rows v+mhalf
    const int jA = j0 + nsrc;
    const int jB = jA + 16;

#pragma unroll
    for (int v = 0; v < 8; ++v) {
      const int s = v + mhalf;
      const float rx1 = __shfl(rb.x, s, 32), ry1 = __shfl(rb.y, s, 32);
      const float rx2 = __shfl(rb.z, s, 32), ry2 = __shfl(rb.w, s, 32);
      const int irow = i0 + s;

      float iw0 = fminf(rx2, c0x2) - fmaxf(rx1, c0x1) + 1.f;
      float ih0 = fminf(ry2, c0y2) - fmaxf(ry1, c0y1) + 1.f;
      float in0 = fmaxf(iw0, 0.f) * fmaxf(ih0, 0.f);
      // iou > T  <=>  inter*(1+T) > T*(area_i+area_j)
      bool p0 = (in0 * (1.f + NMS_T) > NMS_T * c0[v]) && (jA > irow) && (jA < KNMS);

      float iw1 = fminf(rx2, c1x2) - fmaxf(rx1, c1x1) + 1.f;
      float ih1 = fminf(ry2, c1y2) - fmaxf(ry1, c1y1) + 1.f;
      float in1 = fmaxf(iw1, 0.f) * fmaxf(ih1, 0.f);
      bool p1 = (in1 * (1.f + NMS_T) > NMS_T * c1[v]) && (jB > irow) && (jB < KNMS);

      uint32_t b0 = ballot32(p0);
      uint32_t b1 = ballot32(p1);
      if ((lane & 7) == v && lane < 16) {
        uint32_t w = (lane < 8)
            ? ((b0 & 0xFFFFu) | ((b1 & 0xFFFFu) << 16))
            : ((b0 >> 16) | ((b1 >> 16) << 16));
        mask[(size_t)(i0 + lane) * WORDS + wc] = w;
      }
    }
  }
}

// ---------------- 5) sequential greedy scan ----------------
__global__ void nms_scan_kernel(const uint32_t* __restrict__ mask,
                                uint32_t* __restrict__ keepw) {
  __shared__ uint32_t kw[WORDS];
  __shared__ int cur;
  int t = threadIdx.x;
  for (int w = t; w < WORDS; w += blockDim.x) {
    uint32_t m = 0xFFFFFFFFu;
    int rem = KNMS - w * 32;
    if (rem < 32) m = (rem <= 0) ? 0u : ((1u << rem) - 1u);
    kw[w] = m;
  }
  __syncthreads();
  for (int i = 0; i < KNMS; ++i) {
    if (t == 0) cur = (kw[i >> 5] >> (i & 31)) & 1u;
    __syncthreads();
    if (cur) {
      for (int w = t; w < WORDS; w += blockDim.x)
        kw[w] &= ~mask[(size_t)i * WORDS + w];
    }
    __syncthreads();
  }
  for (int w = t; w < WORDS; w += blockDim.x) keepw[w] = kw[w];
}

// ---------------- 6) finalize: emulate top_k tie behavior, write blob ----------------
__global__ void finalize_kernel(const float* __restrict__ topb,
                                const float* __restrict__ tops,
                                const uint32_t* __restrict__ keepw,
                                float* __restrict__ out) {
  __shared__ int ord[POST];
  if (threadIdx.x == 0) {
    int c = 0;
    for (int i = 0; i < KNMS && c < POST; ++i) {
      bool kept = (keepw[i >> 5] >> (i & 31)) & 1u;
      if (kept && tops[i] != -INFINITY) ord[c++] = i;
    }
    for (int i = 0; i < KNMS && c < POST; ++i) {
      bool kept = (keepw[i >> 5] >> (i & 31)) & 1u;
      if (!(kept && tops[i] != -INFINITY)) ord[c++] = i;
    }
    while (c < POST) ord[c++] = 0;
  }
  __syncthreads();
  for (int r = threadIdx.x; r < POST; r += blockDim.x) {
    int i = ord[r];
    out[r * 5 + 0] = 0.f;
    out[r * 5 + 1] = topb[i * 4 + 0];
    out[r * 5 + 2] = topb[i * 4 + 1];
    out[r * 5 + 3] = topb[i * 4 + 2];
    out[r * 5 + 4] = topb[i * 4 + 3];
  }
}

extern "C" void kernel_launch(void* const* d_in, const int* in_sizes, int n_in,
                              void* d_out, int out_size, void* d_ws, size_t ws_size,
                              hipStream_t stream) {
  const float* cls = (const float*)d_in[0];   // [1,64,64,18]
  const float* bb  = (const float*)d_in[1];   // [1,64,64,36]
  float* out = (float*)d_out;                 // [300,5]
  char* ws = (char*)d_ws;                     // needs ~5.75 MB

  float*    boxes = (float*)(ws + OFF_BOXES);
  float*    key   = (float*)(ws + OFF_KEY);
  int*      val   = (int*)(ws + OFF_VAL);
  float*    topb  = (float*)(ws + OFF_TOPB);
  float*    tops  = (float*)(ws + OFF_TOPS);
  uint32_t* mask  = (uint32_t*)(ws + OFF_MASK);
  uint32_t* keepw = (uint32_t*)(ws + OFF_KEEPW);

  rpn_decode_kernel<<<SORT_N / 256, 256, 0, stream>>>(cls, bb, boxes, key, val);

  for (int k = 2; k <= SORT_N; k <<= 1)
    for (int j = k >> 1; j > 0; j >>= 1)
      bitonic_kernel<<<SORT_N / 256, 256, 0, stream>>>(key, val, j, k);

  gather_kernel<<<(KPAD + 255) / 256, 256, 0, stream>>>(key, val, boxes, topb, tops);

  int waves = ROWT * WORDS;                 // 70500 tiles, one wave32 each
  int blocks = (waves + 7) / 8;             // 8 waves per 256-thread block
  nms_mask_kernel<<<blocks, 256, 0, stream>>>(topb, mask);

  nms_scan_kernel<<<1, 256, 0, stream>>>(mask, keepw);
  finalize_kernel<<<1, 256, 0, stream>>>(topb, tops, keepw, out);
}